// Selfattention_78941498900659
// MI455X (gfx1250) — compile-verified
//
#include <hip/hip_runtime.h>
#include <math.h>

// Problem constants (from reference)
#define B_   32
#define C_   1024
#define H_   8
#define IN_  8192            // C*H
#define OUT_ 8192
#define NT_  (OUT_ / 16)     // 512 N-tiles per matrix
#define SCALE_ 0.35355339059327373f  // 1/sqrt(8)

typedef __attribute__((ext_vector_type(2))) float v2f;
typedef __attribute__((ext_vector_type(8))) float v8f;

// ---------------------------------------------------------------------------
// Kernel 1: fused QKV projection.  One wave32 per (matrix, 16-col N-tile).
// D = X * W^T + bias, with X [32 x 8192] fp32, W [8192 x 8192] fp32 row-major.
// Uses V_WMMA_F32_16X16X4_F32: A = X tile (16x4), B = W^T tile (4x16),
// two M-tiles (batch rows 0..15 / 16..31) share each weight load.
// ---------------------------------------------------------------------------
__global__ __launch_bounds__(256) void qkv_gemm_wmma(
    const float* __restrict__ X,
    const float* __restrict__ Wq, const float* __restrict__ bq,
    const float* __restrict__ Wk, const float* __restrict__ bk,
    const float* __restrict__ Wv, const float* __restrict__ bv,
    float* __restrict__ qbuf, float* __restrict__ kbuf, float* __restrict__ vbuf)
{
    const int wave = threadIdx.x >> 5;
    const int t    = blockIdx.x * 8 + wave;   // global wave id, 0..1535
    const int sel  = t >> 9;                  // 0=Q 1=K 2=V
    const int ntil = t & (NT_ - 1);

    const float* W; const float* bias; float* out;
    if (sel == 0)      { W = Wq; bias = bq; out = qbuf; }
    else if (sel == 1) { W = Wk; bias = bk; out = kbuf; }
    else               { W = Wv; bias = bv; out = vbuf; }

    const int l  = threadIdx.x & 31;
    const int lm = l & 15;        // col-in-tile (B/C/D) or row (A)
    const int lh = l >> 4;        // half-wave: k offset +2

    // B operand: lane lm holds W[ntil*16+lm, k + 2*lh ... +1]  (4x16 K-major)
    const float* bp  = W + (size_t)(ntil * 16 + lm) * IN_ + 2 * lh;
    // A operands: lane lm holds X[row, k + 2*lh ... +1]        (16x4)
    const float* ap0 = X + (size_t)lm        * IN_ + 2 * lh;   // batches 0..15
    const float* ap1 = X + (size_t)(lm + 16) * IN_ + 2 * lh;   // batches 16..31

    v8f acc0 = {};
    v8f acc1 = {};

    for (int kk = 0; kk < IN_; kk += 64) {
        // stream the weight rows ahead (global_prefetch_b8)
        if (kk + 512 < IN_) __builtin_prefetch(bp + kk + 512, 0, 1);
#pragma unroll
        for (int k4 = 0; k4 < 64; k4 += 4) {
            const int k = kk + k4;
            v2f a0 = *(const v2f*)(ap0 + k);
            v2f a1 = *(const v2f*)(ap1 + k);
            v2f bb = *(const v2f*)(bp + k);
            acc0 = __builtin_amdgcn_wmma_f32_16x16x4_f32(
                false, a0, false, bb, (short)0, acc0, false, false);
            acc1 = __builtin_amdgcn_wmma_f32_16x16x4_f32(
                false, a1, false, bb, (short)0, acc1, false, false);
        }
    }

    // D layout: vgpr v, lane l -> row (v + 8*lh), col lm
    const float  bval = bias[ntil * 16 + lm];
    const size_t col  = (size_t)ntil * 16 + lm;
#pragma unroll
    for (int v = 0; v < 8; ++v) {
        const int m = v + 8 * lh;
        out[(size_t)m        * OUT_ + col] = acc0[v] + bval;
        out[(size_t)(m + 16) * OUT_ + col] = acc1[v] + bval;
    }
}

// ---------------------------------------------------------------------------
// Kernel 2: per-row softmax stats.  s[e,c] = SCALE * q[e].k[c]; over c=0..1023
// compute m_e = max, rl_e = 1/sum(exp(s-m)).  One thread per row e.
// K rows of this batch live in LDS (uniform-address broadcast reads).
// ---------------------------------------------------------------------------
__global__ __launch_bounds__(256) void attn_row_stats(
    const float* __restrict__ qbuf, const float* __restrict__ kbuf,
    float* __restrict__ mbuf, float* __restrict__ rlbuf)
{
    __shared__ __align__(16) float Ks[C_ * H_];   // 32 KB

    const int b     = blockIdx.x >> 2;
    const int chunk = blockIdx.x & 3;

    const float4* src = (const float4*)(kbuf + (size_t)b * OUT_);
    float4*       dst = (float4*)Ks;
    for (int i = threadIdx.x; i < (C_ * H_) / 4; i += 256) dst[i] = src[i];
    __syncthreads();

    const int    e  = chunk * 256 + threadIdx.x;
    const float* qr = qbuf + (size_t)b * OUT_ + (size_t)e * H_;
    const float4 qa = *(const float4*)(qr);
    const float4 qb = *(const float4*)(qr + 4);

    float mx = -3.0e38f;
    for (int c = 0; c < C_; ++c) {
        const float4 ka = *(const float4*)(Ks + c * H_);
        const float4 kb = *(const float4*)(Ks + c * H_ + 4);
        float s = qa.x * ka.x + qa.y * ka.y + qa.z * ka.z + qa.w * ka.w +
                  qb.x * kb.x + qb.y * kb.y + qb.z * kb.z + qb.w * kb.w;
        s *= SCALE_;
        mx = fmaxf(mx, s);
    }
    float lsum = 0.0f;
    for (int c = 0; c < C_; ++c) {
        const float4 ka = *(const float4*)(Ks + c * H_);
        const float4 kb = *(const float4*)(Ks + c * H_ + 4);
        float s = qa.x * ka.x + qa.y * ka.y + qa.z * ka.z + qa.w * ka.w +
                  qb.x * kb.x + qb.y * kb.y + qb.z * kb.z + qb.w * kb.w;
        s *= SCALE_;
        lsum += __expf(s - mx);
    }
    mbuf[b * C_ + e]  = mx;
    rlbuf[b * C_ + e] = 1.0f / lsum;
}

// ---------------------------------------------------------------------------
// Kernel 3: out[b,c,h] = sum_e exp(s[e,c]-m_e)*rl_e * v[e,h]  (attn^T @ V).
// One thread per output column c; Q, V and stats in LDS (uniform broadcast),
// k[c] held in registers; recompute the 8-wide dot instead of materializing
// the 128 MB score matrix.
// ---------------------------------------------------------------------------
__global__ __launch_bounds__(256) void attn_out(
    const float* __restrict__ qbuf, const float* __restrict__ kbuf,
    const float* __restrict__ vbuf, const float* __restrict__ mbuf,
    const float* __restrict__ rlbuf, float* __restrict__ out)
{
    __shared__ __align__(16) float Qs[C_ * H_];   // 32 KB
    __shared__ __align__(16) float Vs[C_ * H_];   // 32 KB
    __shared__ float Ms[C_];                      // 4 KB
    __shared__ float Ls[C_];                      // 4 KB

    const int b     = blockIdx.x >> 2;
    const int chunk = blockIdx.x & 3;

    {
        const float4* qs = (const float4*)(qbuf + (size_t)b * OUT_);
        const float4* vs = (const float4*)(vbuf + (size_t)b * OUT_);
        float4* qd = (float4*)Qs;
        float4* vd = (float4*)Vs;
        for (int i = threadIdx.x; i < (C_ * H_) / 4; i += 256) {
            qd[i] = qs[i];
            vd[i] = vs[i];
        }
        for (int i = threadIdx.x; i < C_; i += 256) {
            Ms[i] = mbuf[b * C_ + i];
            Ls[i] = rlbuf[b * C_ + i];
        }
    }
    __syncthreads();

    const int    c  = chunk * 256 + threadIdx.x;
    const float* kr = kbuf + (size_t)b * OUT_ + (size_t)c * H_;
    const float4 ka = *(const float4*)(kr);
    const float4 kb = *(const float4*)(kr + 4);

    float4 accA = {0.f, 0.f, 0.f, 0.f};
    float4 accB = {0.f, 0.f, 0.f, 0.f};

    for (int e = 0; e < C_; ++e) {
        const float4 qa = *(const float4*)(Qs + e * H_);
        const float4 qb = *(const float4*)(Qs + e * H_ + 4);
        float s = qa.x * ka.x + qa.y * ka.y + qa.z * ka.z + qa.w * ka.w +
                  qb.x * kb.x + qb.y * kb.y + qb.z * kb.z + qb.w * kb.w;
        s *= SCALE_;
        const float p = __expf(s - Ms[e]) * Ls[e];
        const float4 va = *(const float4*)(Vs + e * H_);
        const float4 vb = *(const float4*)(Vs + e * H_ + 4);
        accA.x += p * va.x; accA.y += p * va.y;
        accA.z += p * va.z; accA.w += p * va.w;
        accB.x += p * vb.x; accB.y += p * vb.y;
        accB.z += p * vb.z; accB.w += p * vb.w;
    }

    float* orow = out + (size_t)b * C_ * H_ + (size_t)c * H_;
    *(float4*)(orow)     = accA;
    *(float4*)(orow + 4) = accB;
}

// ---------------------------------------------------------------------------
extern "C" void kernel_launch(void* const* d_in, const int* in_sizes, int n_in,
                              void* d_out, int out_size, void* d_ws, size_t ws_size,
                              hipStream_t stream)
{
    (void)in_sizes; (void)n_in; (void)out_size; (void)ws_size;

    const float* x  = (const float*)d_in[0];
    const float* Wq = (const float*)d_in[1];
    const float* bq = (const float*)d_in[2];
    const float* Wk = (const float*)d_in[3];
    const float* bk = (const float*)d_in[4];
    const float* Wv = (const float*)d_in[5];
    const float* bv = (const float*)d_in[6];
    float* out = (float*)d_out;

    // workspace layout (floats): qbuf | kbuf | vbuf | mbuf | rlbuf  (~3.3 MB)
    float* qbuf  = (float*)d_ws;
    float* kbuf  = qbuf + (size_t)B_ * OUT_;
    float* vbuf  = kbuf + (size_t)B_ * OUT_;
    float* mbuf  = vbuf + (size_t)B_ * OUT_;
    float* rlbuf = mbuf + (size_t)B_ * C_;

    // 1536 waves = 3 matrices x 512 N-tiles; 8 waves per 256-thread block
    qkv_gemm_wmma<<<192, 256, 0, stream>>>(x, Wq, bq, Wk, bk, Wv, bv,
                                           qbuf, kbuf, vbuf);
    // 32 batches x 4 row-chunks
    attn_row_stats<<<128, 256, 0, stream>>>(qbuf, kbuf, mbuf, rlbuf);
    attn_out<<<128, 256, 0, stream>>>(qbuf, kbuf, vbuf, mbuf, rlbuf, out);
}